// PENN_16836271800590
// MI455X (gfx1250) — compile-verified
//
#include <hip/hip_runtime.h>
#include <hip/hip_bf16.h>

typedef float v2f __attribute__((ext_vector_type(2)));
typedef float v8f __attribute__((ext_vector_type(8)));

#define BB 16
#define NN 512
#define FF 4
#define HH 32
#define BN (BB * NN)

// ---------- WMMA helpers (V_WMMA_F32_16X16X4_F32, full f32 precision) ----------

static __device__ __forceinline__ v8f wmma4(v2f a, v2f b, v8f c) {
    // 8 args: (neg_a, A, neg_b, B, c_mod, C, reuse_a, reuse_b)
    return __builtin_amdgcn_wmma_f32_16x16x4_f32(false, a, false, b, (short)0, c,
                                                 false, false);
}

// A fragment: 16x4 tile of row-major A (leading dim ldm), rows m=lane&15,
// lane group g picks K columns {k0+2g, k0+2g+1}.
static __device__ __forceinline__ v2f ldA(const float* A, int ldm, int m, int k0, int g) {
    const float* q = A + (size_t)m * ldm + k0 + 2 * g;
    v2f r; r.x = q[0]; r.y = q[1];
    return r;
}

// B fragment: 4x16 tile of row-major W[K x 32], column half n0 (0 or 16).
static __device__ __forceinline__ v2f ldB32(const float* W, int k0, int n0, int ln15, int g) {
    v2f r;
    r.x = W[(k0 + 2 * g) * HH + n0 + ln15];
    r.y = W[(k0 + 2 * g + 1) * HH + n0 + ln15];
    return r;
}

static __device__ __forceinline__ v8f bias8(const float* b, int col) {
    float x = b[col];
    v8f r;
#pragma unroll
    for (int v = 0; v < 8; ++v) r[v] = x;
    return r;
}

static __device__ __forceinline__ v8f lrelu8(v8f x) {
#pragma unroll
    for (int v = 0; v < 8; ++v) x[v] = fmaxf(x[v], 0.01f * x[v]);
    return x;
}

// ---------- Kernel A: per-node layer-0 projections + zero alpha accumulator ----------
// c_psi = x@Wsrc + b0 ; d_psi = x@Wdst  (rows 0-3 / 6-9 of psi_w0)
// c_phi = x@Wsrc + b0 ; d_phi = x@Wdst  (rows 0-3 / 5-8 of phi_w0)
__global__ __launch_bounds__(256) void PENN_node_pre(
    const float* __restrict__ node_feat,
    const float* __restrict__ psi_w0, const float* __restrict__ psi_b0,
    const float* __restrict__ phi_w0, const float* __restrict__ phi_b0,
    float* __restrict__ c_psi, float* __restrict__ d_psi,
    float* __restrict__ c_phi, float* __restrict__ d_phi,
    float* __restrict__ ha_sum)
{
    int t = blockIdx.x * 256 + threadIdx.x;
    if (t < BB * HH) ha_sum[t] = 0.f;
    if (t >= BN * HH) return;
    int bn = t >> 5, k = t & 31;
    float x0 = node_feat[bn * 4 + 0], x1 = node_feat[bn * 4 + 1];
    float x2 = node_feat[bn * 4 + 2], x3 = node_feat[bn * 4 + 3];
    c_psi[t] = psi_b0[k] + x0 * psi_w0[0 * HH + k] + x1 * psi_w0[1 * HH + k]
                         + x2 * psi_w0[2 * HH + k] + x3 * psi_w0[3 * HH + k];
    d_psi[t] =             x0 * psi_w0[6 * HH + k] + x1 * psi_w0[7 * HH + k]
                         + x2 * psi_w0[8 * HH + k] + x3 * psi_w0[9 * HH + k];
    c_phi[t] = phi_b0[k] + x0 * phi_w0[0 * HH + k] + x1 * phi_w0[1 * HH + k]
                         + x2 * phi_w0[2 * HH + k] + x3 * phi_w0[3 * HH + k];
    d_phi[t] =             x0 * phi_w0[5 * HH + k] + x1 * phi_w0[6 * HH + k]
                         + x2 * phi_w0[7 * HH + k] + x3 * phi_w0[8 * HH + k];
}

// ---------- Kernel B: edge accumulate (dominant stage) ----------
// For each node i: s = (1/N) * sum_j lrelu(c_i + d_j + Lij*wL (+ Lji*wLt))
// Block = 8 nodes x 32 channels; j tiled by 32 through LDS.
__global__ __launch_bounds__(256) void PENN_edge(
    const float* __restrict__ liab,
    const float* __restrict__ c_psi, const float* __restrict__ d_psi,
    const float* __restrict__ c_phi, const float* __restrict__ d_phi,
    const float* __restrict__ psi_w0, const float* __restrict__ phi_w0,
    float* __restrict__ s_psi, float* __restrict__ s_phi)
{
    __shared__ float sdp[32][32];
    __shared__ float sdf[32][32];
    __shared__ float sl[8][32];
    __shared__ float slt[32][8];
    const int k = threadIdx.x;       // channel 0..31 (lane)
    const int ii = threadIdx.y;      // node-in-block 0..7 (wave)
    const int tid = ii * 32 + k;
    const int bnBase = blockIdx.x * 8;        // never straddles a batch (512 % 8 == 0)
    const int bn = bnBase + ii;
    const int b = bn >> 9;
    const float* dp = d_psi + (size_t)b * NN * HH;
    const float* df = d_phi + (size_t)b * NN * HH;
    const float wl  = psi_w0[4 * HH + k];     // L_ij row of psi_w0
    const float wlt = psi_w0[5 * HH + k];     // L_ji row of psi_w0
    const float wlp = phi_w0[4 * HH + k];     // L_ij row of phi_w0
    const float cps = c_psi[(size_t)bn * HH + k];
    const float cph = c_phi[(size_t)bn * HH + k];
    float accp = 0.f, accf = 0.f;
    // cooperative L^T sub-block load mapping
    const int jj2 = tid >> 3, c2 = tid & 7;
    const int bn2 = bnBase + c2;
    const size_t colBase = (size_t)(bn2 >> 9) * NN * NN + (size_t)(bn2 & (NN - 1));

    for (int j0 = 0; j0 < NN; j0 += 32) {
        __syncthreads();
        // stage 32 rows of d_psi/d_phi (shared by all 8 nodes in block)
#pragma unroll
        for (int t = 0; t < 4; ++t) {
            int e = tid + t * 256;
            int jj = e >> 5, kk = e & 31;
            sdp[jj][kk] = dp[(size_t)(j0 + jj) * HH + kk];
            sdf[jj][kk] = df[(size_t)(j0 + jj) * HH + kk];
        }
        // L rows (coalesced) and L^T column sub-block (32j x 8i)
        sl[ii][k] = liab[(size_t)bn * NN + j0 + k];
        slt[jj2][c2] = liab[colBase + (size_t)(j0 + jj2) * NN];
        if (j0 + 32 < NN) {
            __builtin_prefetch(&dp[(size_t)(j0 + 32 + (tid >> 5)) * HH + (tid & 31)], 0, 3);
            __builtin_prefetch(&df[(size_t)(j0 + 32 + (tid >> 5)) * HH + (tid & 31)], 0, 3);
        }
        __syncthreads();
#pragma unroll 4
        for (int jj = 0; jj < 32; ++jj) {
            float Lij = sl[ii][jj];
            float Lji = slt[jj][ii];
            float zp = fmaf(Lji, wlt, fmaf(Lij, wl, cps + sdp[jj][k]));
            accp += fmaxf(zp, 0.01f * zp);
            float zf = fmaf(Lij, wlp, cph + sdf[jj][k]);
            accf += fmaxf(zf, 0.01f * zf);
        }
    }
    s_psi[(size_t)bn * HH + k] = accp * (1.0f / (float)NN);
    s_phi[(size_t)bn * HH + k] = accf * (1.0f / (float)NN);
}

// ---------- Kernel C: WMMA node stage ----------
// h_psi = s_psi@psi_w1 + b ; h_phi = s_phi@phi_w1 + b (kept in regs/LDS);
// alpha MLP on [x | h_phi]; per-batch column-sum into ha_sum via atomics.
__global__ __launch_bounds__(32) void PENN_node_stage(
    const float* __restrict__ node_feat,
    const float* __restrict__ s_psi, const float* __restrict__ s_phi,
    const float* __restrict__ psi_w1, const float* __restrict__ psi_b1,
    const float* __restrict__ phi_w1, const float* __restrict__ phi_b1,
    const float* __restrict__ alpha_w0, const float* __restrict__ alpha_b0,
    const float* __restrict__ alpha_w1, const float* __restrict__ alpha_b1,
    float* __restrict__ h_psi, float* __restrict__ ha_sum)
{
    const int lane = threadIdx.x;
    const int ln15 = lane & 15;
    const int g = lane >> 4;
    const int base = blockIdx.x * 16;
    const int b = base >> 9;
    __shared__ float sh[16][33];
    __shared__ float sh2[16][33];

    // psi layer 1: [16x32] @ [32x32] + b1
    v8f pl = bias8(psi_b1, ln15);
    v8f ph = bias8(psi_b1, 16 + ln15);
#pragma unroll
    for (int k0 = 0; k0 < 32; k0 += 4) {
        v2f a = ldA(s_psi + (size_t)base * HH, HH, ln15, k0, g);
        pl = wmma4(a, ldB32(psi_w1, k0, 0, ln15, g), pl);
        ph = wmma4(a, ldB32(psi_w1, k0, 16, ln15, g), ph);
    }
#pragma unroll
    for (int v = 0; v < 8; ++v) {
        int row = base + v + 8 * g;
        h_psi[(size_t)row * HH + ln15] = pl[v];
        h_psi[(size_t)row * HH + 16 + ln15] = ph[v];
    }

    // phi layer 1 -> stage to LDS (C-layout -> A-layout conversion)
    v8f fl = bias8(phi_b1, ln15);
    v8f fh = bias8(phi_b1, 16 + ln15);
#pragma unroll
    for (int k0 = 0; k0 < 32; k0 += 4) {
        v2f a = ldA(s_phi + (size_t)base * HH, HH, ln15, k0, g);
        fl = wmma4(a, ldB32(phi_w1, k0, 0, ln15, g), fl);
        fh = wmma4(a, ldB32(phi_w1, k0, 16, ln15, g), fh);
    }
#pragma unroll
    for (int v = 0; v < 8; ++v) {
        sh[v + 8 * g][ln15] = fl[v];
        sh[v + 8 * g][16 + ln15] = fh[v];
    }
    __syncthreads();

    // alpha layer 0: A = [x | h_phi], K = 36
    v8f al = bias8(alpha_b0, ln15);
    v8f ah = bias8(alpha_b0, 16 + ln15);
    {
        v2f a = ldA(node_feat + (size_t)base * FF, FF, ln15, 0, g);
        al = wmma4(a, ldB32(alpha_w0, 0, 0, ln15, g), al);
        ah = wmma4(a, ldB32(alpha_w0, 0, 16, ln15, g), ah);
    }
#pragma unroll
    for (int k0 = 4; k0 < 36; k0 += 4) {
        v2f a;
        a.x = sh[ln15][k0 - 4 + 2 * g];
        a.y = sh[ln15][k0 - 3 + 2 * g];
        al = wmma4(a, ldB32(alpha_w0, k0, 0, ln15, g), al);
        ah = wmma4(a, ldB32(alpha_w0, k0, 16, ln15, g), ah);
    }
    al = lrelu8(al);
    ah = lrelu8(ah);
#pragma unroll
    for (int v = 0; v < 8; ++v) {
        sh2[v + 8 * g][ln15] = al[v];
        sh2[v + 8 * g][16 + ln15] = ah[v];
    }
    __syncthreads();

    // alpha layer 1: K = 32
    v8f ql = bias8(alpha_b1, ln15);
    v8f qh = bias8(alpha_b1, 16 + ln15);
#pragma unroll
    for (int k0 = 0; k0 < 32; k0 += 4) {
        v2f a;
        a.x = sh2[ln15][k0 + 2 * g];
        a.y = sh2[ln15][k0 + 1 + 2 * g];
        ql = wmma4(a, ldB32(alpha_w1, k0, 0, ln15, g), ql);
        qh = wmma4(a, ldB32(alpha_w1, k0, 16, ln15, g), qh);
    }

    // column sums over the tile's 16 rows -> batch accumulator
    float slo = 0.f, shi = 0.f;
#pragma unroll
    for (int v = 0; v < 8; ++v) { slo += ql[v]; shi += qh[v]; }
    slo += __shfl_xor(slo, 16, 32);
    shi += __shfl_xor(shi, 16, 32);
    float contrib = (lane < 16) ? slo : shi;
    atomicAdd(&ha_sum[b * HH + lane], contrib);
}

// ---------- Kernel D: rho stage (WMMA, K = 68 = [x(4) | h_alpha(32) | h_psi(32)]) ----------
__global__ __launch_bounds__(32) void PENN_rho(
    const float* __restrict__ node_feat,
    const float* __restrict__ h_psi,
    const float* __restrict__ ha_sum,
    const float* __restrict__ rho_w0, const float* __restrict__ rho_b0,
    const float* __restrict__ rho_w1, const float* __restrict__ rho_b1,
    float* __restrict__ out)
{
    const int lane = threadIdx.x;
    const int ln15 = lane & 15;
    const int g = lane >> 4;
    const int base = blockIdx.x * 16;
    const int b = base >> 9;
    __shared__ float sha[32];
    __shared__ float sh[16][33];
    sha[lane] = ha_sum[b * HH + lane] * (1.0f / (float)NN);
    __syncthreads();

    v8f rl = bias8(rho_b0, ln15);
    v8f rh = bias8(rho_b0, 16 + ln15);
    {
        v2f a = ldA(node_feat + (size_t)base * FF, FF, ln15, 0, g);
        rl = wmma4(a, ldB32(rho_w0, 0, 0, ln15, g), rl);
        rh = wmma4(a, ldB32(rho_w0, 0, 16, ln15, g), rh);
    }
#pragma unroll
    for (int k0 = 4; k0 < 36; k0 += 4) {    // h_alpha columns (row-invariant broadcast)
        v2f a;
        a.x = sha[k0 - 4 + 2 * g];
        a.y = sha[k0 - 3 + 2 * g];
        rl = wmma4(a, ldB32(rho_w0, k0, 0, ln15, g), rl);
        rh = wmma4(a, ldB32(rho_w0, k0, 16, ln15, g), rh);
    }
#pragma unroll
    for (int k0 = 36; k0 < 68; k0 += 4) {   // h_psi columns
        v2f a = ldA(h_psi + (size_t)base * HH, HH, ln15, k0 - 36, g);
        rl = wmma4(a, ldB32(rho_w0, k0, 0, ln15, g), rl);
        rh = wmma4(a, ldB32(rho_w0, k0, 16, ln15, g), rh);
    }
    rl = lrelu8(rl);
    rh = lrelu8(rh);
#pragma unroll
    for (int v = 0; v < 8; ++v) {
        sh[v + 8 * g][ln15] = rl[v];
        sh[v + 8 * g][16 + ln15] = rh[v];
    }
    __syncthreads();
    if (lane < 16) {
        float o = rho_b1[0];
#pragma unroll
        for (int c = 0; c < 32; ++c) o += sh[lane][c] * rho_w1[c];
        out[base + lane] = o;
    }
}

// ---------- launch ----------
extern "C" void kernel_launch(void* const* d_in, const int* in_sizes, int n_in,
                              void* d_out, int out_size, void* d_ws, size_t ws_size,
                              hipStream_t stream) {
    const float* node_feat = (const float*)d_in[0];
    const float* liab      = (const float*)d_in[1];
    const float* psi_w0    = (const float*)d_in[2];
    const float* psi_b0    = (const float*)d_in[3];
    const float* psi_w1    = (const float*)d_in[4];
    const float* psi_b1    = (const float*)d_in[5];
    const float* phi_w0    = (const float*)d_in[6];
    const float* phi_b0    = (const float*)d_in[7];
    const float* phi_w1    = (const float*)d_in[8];
    const float* phi_b1    = (const float*)d_in[9];
    const float* alpha_w0  = (const float*)d_in[10];
    const float* alpha_b0  = (const float*)d_in[11];
    const float* alpha_w1  = (const float*)d_in[12];
    const float* alpha_b1  = (const float*)d_in[13];
    const float* rho_w0    = (const float*)d_in[14];
    const float* rho_b0    = (const float*)d_in[15];
    const float* rho_w1    = (const float*)d_in[16];
    const float* rho_b1    = (const float*)d_in[17];

    float* ws = (float*)d_ws;
    const size_t NH = (size_t)BN * HH;
    float* c_psi  = ws;
    float* d_psi_ = c_psi + NH;
    float* c_phi  = d_psi_ + NH;
    float* d_phi_ = c_phi + NH;
    float* s_psi  = d_phi_ + NH;
    float* s_phi  = s_psi + NH;
    float* h_psi  = s_phi + NH;
    float* ha_sum = h_psi + NH;          // BB*HH = 512 floats
    float* out = (float*)d_out;

    PENN_node_pre<<<(BN * HH + 255) / 256, 256, 0, stream>>>(
        node_feat, psi_w0, psi_b0, phi_w0, phi_b0,
        c_psi, d_psi_, c_phi, d_phi_, ha_sum);

    PENN_edge<<<BN / 8, dim3(32, 8), 0, stream>>>(
        liab, c_psi, d_psi_, c_phi, d_phi_, psi_w0, phi_w0, s_psi, s_phi);

    PENN_node_stage<<<BN / 16, 32, 0, stream>>>(
        node_feat, s_psi, s_phi, psi_w1, psi_b1, phi_w1, phi_b1,
        alpha_w0, alpha_b0, alpha_w1, alpha_b1, h_psi, ha_sum);

    PENN_rho<<<BN / 16, 32, 0, stream>>>(
        node_feat, h_psi, ha_sum, rho_w0, rho_b0, rho_w1, rho_b1, out);
}